// ModelWithEdgeFeatures_49555332661695
// MI455X (gfx1250) — compile-verified
//
#include <hip/hip_runtime.h>
#include <hip/hip_bf16.h>

#define N_NODES   100000
#define N_EDGES   800000
#define N_GRAPHS  1024
#define HID       128
#define N_LAYERS  3
#define EDGE_CH   16
#define MLP_HID   256
#define N_CLASSES 10
#define BN_EPS    1e-5f

typedef float v2f __attribute__((ext_vector_type(2)));
typedef float v8f __attribute__((ext_vector_type(8)));

// ---------------------------------------------------------------- utilities
__global__ void zero_v4(float4* __restrict__ p, long long n4) {
    long long i = blockIdx.x * (long long)blockDim.x + threadIdx.x;
    long long stride = (long long)gridDim.x * blockDim.x;
    float4 z = {0.f, 0.f, 0.f, 0.f};
    for (; i < n4; i += stride) p[i] = z;
}

// ------------------------------------------- layer-invariant edge aggregates
// ES[v] = sum_{e: dst=v} edge_attr[e]   (N x 16),  deg[v] = #incoming edges
__global__ void edge_static_scatter(const int* __restrict__ dst,
                                    const float* __restrict__ eattr,
                                    float* __restrict__ ES,
                                    float* __restrict__ deg) {
    long long t = blockIdx.x * (long long)blockDim.x + threadIdx.x;
    if (t >= (long long)N_EDGES * EDGE_CH) return;
    int e = (int)(t >> 4), c = (int)(t & 15);
    int d = dst[e];
    atomicAdd(&ES[(long long)d * EDGE_CH + c], eattr[(long long)e * EDGE_CH + c]);
    if (c == 0) atomicAdd(&deg[d], 1.0f);
}

// ---------------------------------------------------- per-layer edge scatter
// One wave per edge: lane handles 4 channels (global_load_b128 gather).
// Fuses the previous layer's BN affine + ReLU into the gather, so the
// normalized activations are never materialized:  S[dst] += relu(aff(hin[src]))
__global__ void edge_scatter_fused(const int* __restrict__ src,
                                   const int* __restrict__ dst,
                                   const float* __restrict__ hin,
                                   const float* __restrict__ scale,
                                   const float* __restrict__ shift,
                                   int affine,
                                   float* __restrict__ S) {
    const int e    = blockIdx.x * 8 + (threadIdx.x >> 5);  // 8 waves / block
    const int lane = threadIdx.x & 31;
    const int s = src[e];      // wave-uniform
    const int d = dst[e];      // wave-uniform
    float4 v = ((const float4*)hin)[(long long)s * 32 + lane];
    if (affine) {
        const int c = lane * 4;
        v.x = fmaxf(0.f, v.x * scale[c + 0] + shift[c + 0]);
        v.y = fmaxf(0.f, v.y * scale[c + 1] + shift[c + 1]);
        v.z = fmaxf(0.f, v.z * scale[c + 2] + shift[c + 2]);
        v.w = fmaxf(0.f, v.w * scale[c + 3] + shift[c + 3]);
    }
    float* p = S + (long long)d * HID + lane * 4;
    atomicAdd(p + 0, v.x);
    atomicAdd(p + 1, v.y);
    atomicAdd(p + 2, v.z);
    atomicAdd(p + 3, v.w);
}

// -------------------------------------------------------- fp32 WMMA GEMM
// agg = S @ Wx_l + ES @ We_l + deg (outer) bmsg_l
// One wave computes one 16x16 tile via V_WMMA_F32_16X16X4_F32 (exact fp32).
__global__ void layer_gemm_wmma(const float* __restrict__ S,
                                const float* __restrict__ ES,
                                const float* __restrict__ deg,
                                const float* __restrict__ Wx,
                                const float* __restrict__ We,
                                const float* __restrict__ bmsg,
                                float* __restrict__ agg) {
    const int wave = threadIdx.x >> 5;
    const int lane = threadIdx.x & 31;
    const int half = lane >> 4;     // 0: K pair {0,1}, 1: K pair {2,3}
    const int l16  = lane & 15;
    const int m0 = blockIdx.x * 16;
    const int n0 = (blockIdx.y * 4 + wave) * 16;
    const int arow = m0 + l16;      // A fragment row (M = lane%16)
    const int bcol = n0 + l16;      // B fragment col (N = lane%16)

    v8f c = {0.f, 0.f, 0.f, 0.f, 0.f, 0.f, 0.f, 0.f};

    const float* Sr = S + (long long)arow * HID;
    #pragma unroll 4
    for (int k0 = 0; k0 < HID; k0 += 4) {
        const int ka = k0 + half * 2;
        v2f a; a.x = Sr[ka];               a.y = Sr[ka + 1];
        v2f b; b.x = Wx[ka * HID + bcol];  b.y = Wx[(ka + 1) * HID + bcol];
        c = __builtin_amdgcn_wmma_f32_16x16x4_f32(false, a, false, b,
                                                  (short)0, c, false, false);
    }
    const float* Er = ES + (long long)arow * EDGE_CH;
    #pragma unroll
    for (int k0 = 0; k0 < EDGE_CH; k0 += 4) {
        const int ka = k0 + half * 2;
        v2f a; a.x = Er[ka];               a.y = Er[ka + 1];
        v2f b; b.x = We[ka * HID + bcol];  b.y = We[(ka + 1) * HID + bcol];
        c = __builtin_amdgcn_wmma_f32_16x16x4_f32(false, a, false, b,
                                                  (short)0, c, false, false);
    }
    // D layout: VGPR r -> rows m0+r (lanes 0-15) and m0+r+8 (lanes 16-31)
    #pragma unroll
    for (int r = 0; r < 8; r++) {
        const int orow = m0 + r + half * 8;
        agg[(long long)orow * HID + bcol] = c[r] + deg[orow] * bmsg[bcol];
    }
}

// ------------------------------------------------------------- batch norm
// float4 column accumulation; 8 row-slices per block reduced through LDS.
__global__ void bn_stats(const float4* __restrict__ agg4,
                         float* __restrict__ sums, float* __restrict__ sumsq) {
    __shared__ float4 ss[256];
    __shared__ float4 sq[256];
    const int t = threadIdx.x;
    const int c4 = t & 31, slice = t >> 5;
    float4 s  = {0.f, 0.f, 0.f, 0.f};
    float4 s2 = {0.f, 0.f, 0.f, 0.f};
    for (int r = blockIdx.x * 8 + slice; r < N_NODES; r += gridDim.x * 8) {
        float4 v = agg4[(long long)r * 32 + c4];
        s.x += v.x; s.y += v.y; s.z += v.z; s.w += v.w;
        s2.x += v.x * v.x; s2.y += v.y * v.y; s2.z += v.z * v.z; s2.w += v.w * v.w;
    }
    ss[t] = s; sq[t] = s2;
    __syncthreads();
    if (t < 32) {
        float4 a = ss[t], b = sq[t];
        #pragma unroll
        for (int k = 1; k < 8; k++) {
            float4 u = ss[t + 32 * k], w = sq[t + 32 * k];
            a.x += u.x; a.y += u.y; a.z += u.z; a.w += u.w;
            b.x += w.x; b.y += w.y; b.z += w.z; b.w += w.w;
        }
        atomicAdd(&sums[t * 4 + 0], a.x);  atomicAdd(&sums[t * 4 + 1], a.y);
        atomicAdd(&sums[t * 4 + 2], a.z);  atomicAdd(&sums[t * 4 + 3], a.w);
        atomicAdd(&sumsq[t * 4 + 0], b.x); atomicAdd(&sumsq[t * 4 + 1], b.y);
        atomicAdd(&sumsq[t * 4 + 2], b.z); atomicAdd(&sumsq[t * 4 + 3], b.w);
    }
}

__global__ void bn_finalize(const float* __restrict__ sums,
                            const float* __restrict__ sumsq,
                            const float* __restrict__ gamma,
                            const float* __restrict__ beta,
                            float* __restrict__ scale, float* __restrict__ shift) {
    const int c = threadIdx.x;  // 128 threads
    const float inv_n = 1.0f / (float)N_NODES;
    float mean = sums[c] * inv_n;
    float var  = sumsq[c] * inv_n - mean * mean;
    float sc   = gamma[c] * rsqrtf(var + BN_EPS);
    scale[c] = sc;
    shift[c] = beta[c] - mean * sc;
}

// ------------------------------------------------------------- graph pooling
// One wave per node; fuses the last layer's BN affine + ReLU.
__global__ void pool_scatter_fused(const int* __restrict__ batch,
                                   const float4* __restrict__ agg4,
                                   const float* __restrict__ scale,
                                   const float* __restrict__ shift,
                                   float* __restrict__ pooled) {
    const int n    = blockIdx.x * 8 + (threadIdx.x >> 5);
    const int lane = threadIdx.x & 31;
    float4 v = agg4[(long long)n * 32 + lane];
    const int c = lane * 4;
    v.x = fmaxf(0.f, v.x * scale[c + 0] + shift[c + 0]);
    v.y = fmaxf(0.f, v.y * scale[c + 1] + shift[c + 1]);
    v.z = fmaxf(0.f, v.z * scale[c + 2] + shift[c + 2]);
    v.w = fmaxf(0.f, v.w * scale[c + 3] + shift[c + 3]);
    float* p = pooled + (long long)batch[n] * HID + c;
    atomicAdd(p + 0, v.x);
    atomicAdd(p + 1, v.y);
    atomicAdd(p + 2, v.z);
    atomicAdd(p + 3, v.w);
}

// ------------------------------------------------------------- MLP head
__global__ void fc1_relu(const float* __restrict__ pooled,
                         const float* __restrict__ nb,
                         const float* __restrict__ w, const float* __restrict__ b,
                         float* __restrict__ out1) {
    int t = blockIdx.x * blockDim.x + threadIdx.x;     // G * MLP_HID
    if (t >= N_GRAPHS * MLP_HID) return;
    int g = t / MLP_HID, j = t % MLP_HID;
    float acc = b[j];
    const float* pr = pooled + (long long)g * HID;
    #pragma unroll 4
    for (int k = 0; k < HID; k++) acc += pr[k] * w[k * MLP_HID + j];
    const float* nr = nb + (long long)g * N_CLASSES;
    #pragma unroll
    for (int k = 0; k < N_CLASSES; k++) acc += nr[k] * w[(HID + k) * MLP_HID + j];
    out1[t] = acc > 0.f ? acc : 0.f;
}

__global__ void fc2_softmax(const float* __restrict__ out1,
                            const float* __restrict__ w, const float* __restrict__ b,
                            float* __restrict__ out) {
    int g = blockIdx.x * blockDim.x + threadIdx.x;
    if (g >= N_GRAPHS) return;
    float logit[N_CLASSES];
    #pragma unroll
    for (int j = 0; j < N_CLASSES; j++) logit[j] = b[j];
    const float* orow = out1 + (long long)g * MLP_HID;
    for (int k = 0; k < MLP_HID; k++) {
        float v = orow[k];
        #pragma unroll
        for (int j = 0; j < N_CLASSES; j++) logit[j] += v * w[k * N_CLASSES + j];
    }
    float m = logit[0];
    #pragma unroll
    for (int j = 1; j < N_CLASSES; j++) m = fmaxf(m, logit[j]);
    float s = 0.f;
    #pragma unroll
    for (int j = 0; j < N_CLASSES; j++) { logit[j] = expf(logit[j] - m); s += logit[j]; }
    float inv = 1.0f / s;
    #pragma unroll
    for (int j = 0; j < N_CLASSES; j++) out[(long long)g * N_CLASSES + j] = logit[j] * inv;
}

// ---------------------------------------------------------------- launcher
extern "C" void kernel_launch(void* const* d_in, const int* in_sizes, int n_in,
                              void* d_out, int out_size, void* d_ws, size_t ws_size,
                              hipStream_t stream) {
    (void)in_sizes; (void)n_in; (void)out_size; (void)ws_size;
    const float* x        = (const float*)d_in[0];
    const int*   eidx     = (const int*)  d_in[1];
    const float* eattr    = (const float*)d_in[2];
    const int*   batch    = (const int*)  d_in[3];
    const float* neighbor = (const float*)d_in[4];
    const float* Wx       = (const float*)d_in[5];
    const float* We       = (const float*)d_in[6];
    const float* bmsg     = (const float*)d_in[7];
    const float* gamma    = (const float*)d_in[8];
    const float* beta     = (const float*)d_in[9];
    const float* fc1_w    = (const float*)d_in[10];
    const float* fc1_b    = (const float*)d_in[11];
    const float* fc2_w    = (const float*)d_in[12];
    const float* fc2_b    = (const float*)d_in[13];
    float* out = (float*)d_out;

    const int* src = eidx;            // edge_index[0]
    const int* dst = eidx + N_EDGES;  // edge_index[1]

    // workspace carve-up (256B aligned slots)
    char* ws = (char*)d_ws;
    size_t off = 0;
    auto take = [&](size_t elems) -> float* {
        float* p = (float*)(ws + off);
        off += ((elems * sizeof(float) + 255) / 256) * 256;
        return p;
    };
    float* S      = take((size_t)N_NODES * HID);
    float* aggA   = take((size_t)N_NODES * HID);
    float* aggB   = take((size_t)N_NODES * HID);
    float* ES     = take((size_t)N_NODES * EDGE_CH);
    float* deg    = take(N_NODES);
    float* sums   = take(HID);
    float* sumsq  = take(HID);
    float* scale  = take(HID);
    float* shift  = take(HID);
    float* pooled = take((size_t)N_GRAPHS * HID);
    float* out1   = take((size_t)N_GRAPHS * MLP_HID);

    // layer-invariant edge aggregates: ES = segsum(edge_attr, dst), deg
    zero_v4<<<1024, 256, 0, stream>>>((float4*)ES, (long long)N_NODES * EDGE_CH / 4);
    zero_v4<<<128, 256, 0, stream>>>((float4*)deg, N_NODES / 4);
    edge_static_scatter<<<(N_EDGES * EDGE_CH + 255) / 256, 256, 0, stream>>>(dst, eattr, ES, deg);

    for (int l = 0; l < N_LAYERS; l++) {
        const float* hin = (l == 0) ? x : ((l & 1) ? aggA : aggB);
        float* aggOut    = (l & 1) ? aggB : aggA;
        // S = segsum(relu(aff(hin))[src], dst)  -- affine fused except layer 0
        zero_v4<<<4096, 256, 0, stream>>>((float4*)S, (long long)N_NODES * HID / 4);
        edge_scatter_fused<<<N_EDGES / 8, 256, 0, stream>>>(src, dst, hin, scale, shift,
                                                            (l != 0), S);
        // agg = S @ Wx_l + ES @ We_l + deg * bmsg_l   (fp32 WMMA)
        dim3 ggrid(N_NODES / 16, 2);
        layer_gemm_wmma<<<ggrid, 128, 0, stream>>>(
            S, ES, deg,
            Wx + (size_t)l * HID * HID,
            We + (size_t)l * EDGE_CH * HID,
            bmsg + (size_t)l * HID, aggOut);
        // batch-norm stats -> per-channel affine for the NEXT consumer
        zero_v4<<<1, 64, 0, stream>>>((float4*)sums, HID / 4);
        zero_v4<<<1, 64, 0, stream>>>((float4*)sumsq, HID / 4);
        bn_stats<<<256, 256, 0, stream>>>((const float4*)aggOut, sums, sumsq);
        bn_finalize<<<1, 128, 0, stream>>>(sums, sumsq,
                                           gamma + (size_t)l * HID,
                                           beta + (size_t)l * HID, scale, shift);
    }

    // pooled = segsum(relu(aff(agg_last)), batch)  -- last BN fused into pooling
    const float* aggLast = aggA;  // layer 2 output
    zero_v4<<<128, 256, 0, stream>>>((float4*)pooled, (long long)N_GRAPHS * HID / 4);
    pool_scatter_fused<<<N_NODES / 8, 256, 0, stream>>>(batch, (const float4*)aggLast,
                                                        scale, shift, pooled);

    // head MLP + softmax
    fc1_relu<<<(N_GRAPHS * MLP_HID + 255) / 256, 256, 0, stream>>>(pooled, neighbor, fc1_w, fc1_b, out1);
    fc2_softmax<<<(N_GRAPHS + 255) / 256, 256, 0, stream>>>(out1, fc2_w, fc2_b, out);
}